// VectorQuantizer_34909494182384
// MI455X (gfx1250) — compile-verified
//
#include <hip/hip_runtime.h>
#include <hip/hip_bf16.h>

typedef __attribute__((ext_vector_type(2))) float v2f;
typedef __attribute__((ext_vector_type(8))) float v8f;

#define K_CODES 512
#define DIM 64
#define PIX_PER_BLOCK 256
#define THREADS 256
#define WROW 68               // padded LDS row stride (floats)
#define N_PIX 131072          // 32*64*64
#define N_BLOCKS (N_PIX / PIX_PER_BLOCK)   // 512
#define Q_OFF 131072          // quantized output offset in d_out
#define Q_ELEMS 8388608       // 32*64*64*64

// LDS carve (floats):
//   w_lds  [512*68]   codebook, row-major padded (bank-conflict free, stride 68)
//   xT_lds [256*68]   x tile, [pixel][channel]
//   wn_lds [512]      ||w_j||^2
//   idx_lds[256] (int)
//   red    [8]        per-wave loss partials

__global__ __launch_bounds__(THREADS)
void vq_main(const float* __restrict__ x, const float* __restrict__ weight,
             float* __restrict__ out, float* __restrict__ partial)
{
    extern __shared__ float smem[];
    float* w_lds   = smem;                               // 512*68
    float* xT_lds  = w_lds + K_CODES * WROW;             // 256*68
    float* wn_lds  = xT_lds + PIX_PER_BLOCK * WROW;      // 512
    int*   idx_lds = (int*)(wn_lds + K_CODES);           // 256
    float* red     = (float*)(idx_lds + PIX_PER_BLOCK);  // 8

    const int t   = threadIdx.x;
    const int p0  = blockIdx.x * PIX_PER_BLOCK;          // flat NHWC pixel base
    const int b   = p0 >> 12;                            // / 4096 (H*W)
    const int hw0 = p0 & 4095;
    const float* xb = x + (size_t)b * (DIM * 4096) + hw0;

    // ---- stage codebook into LDS (coalesced global reads) ----
    for (int i = t; i < K_CODES * DIM; i += THREADS) {
        int code = i >> 6, c = i & 63;
        w_lds[code * WROW + c] = weight[i];
    }
    // ---- stage x tile transposed: [pixel][channel]; each channel segment is
    //      256 contiguous floats in global memory (coalesced) ----
    for (int i = t; i < DIM * PIX_PER_BLOCK; i += THREADS) {
        int c = i >> 8, pl = i & 255;
        xT_lds[pl * WROW + c] = xb[(size_t)c * 4096 + pl];
    }
    __syncthreads();

    // ---- per-code squared norms ----
    for (int code = t; code < K_CODES; code += THREADS) {
        const float* wr = w_lds + code * WROW;
        float s = 0.f;
        #pragma unroll 8
        for (int c = 0; c < DIM; ++c) s += wr[c] * wr[c];
        wn_lds[code] = s;
    }
    __syncthreads();

    const int lane = t & 31;
    const int wave = t >> 5;          // 0..7
    const int tblA = wave * 32;       // M-tile A local pixel base
    const int tblB = tblA + 16;       // M-tile B local pixel base
    const int lm   = lane & 15;
    const int lh   = lane >> 4;       // lane half (0/1)

    // ---- A fragments in registers for both M-tiles:
    //      A[M=lm][K = 4*kk + 2*lh + v] ----
    v2f afA[16], afB[16];
    {
        const float* xrA = xT_lds + (tblA + lm) * WROW + 2 * lh;
        const float* xrB = xT_lds + (tblB + lm) * WROW + 2 * lh;
        #pragma unroll
        for (int kk = 0; kk < 16; ++kk) {
            afA[kk] = *(const v2f*)(xrA + 4 * kk);
            afB[kk] = *(const v2f*)(xrB + 4 * kk);
        }
    }

    float bestA[8], bestB[8];
    int   bidxA[8], bidxB[8];
    #pragma unroll
    for (int r = 0; r < 8; ++r) {
        bestA[r] = 3.4e38f; bidxA[r] = 0;
        bestB[r] = 3.4e38f; bidxB[r] = 0;
    }

    // ---- 16 iterations x (2 N-tiles x 2 M-tiles x 16 K-steps).
    //      Each loaded B fragment feeds 2 wmmas (M-tiles A and B) and the 4
    //      accumulator chains are independent: 4 wmmas per ds wait, 4-way ILP.
    for (int j0 = 0; j0 < K_CODES; j0 += 32) {
        const float* wr0 = w_lds + (j0 + lm) * WROW + 2 * lh;   // B[K][N=code]
        const float* wr1 = wr0 + 16 * WROW;

        v8f accA0 = {}, accA1 = {}, accB0 = {}, accB1 = {};
        #pragma unroll
        for (int kk = 0; kk < 16; ++kk) {
            v2f bf0 = *(const v2f*)(wr0 + 4 * kk);
            v2f bf1 = *(const v2f*)(wr1 + 4 * kk);
            accA0 = __builtin_amdgcn_wmma_f32_16x16x4_f32(
                false, afA[kk], false, bf0, (short)0, accA0, false, false);
            accB0 = __builtin_amdgcn_wmma_f32_16x16x4_f32(
                false, afB[kk], false, bf0, (short)0, accB0, false, false);
            accA1 = __builtin_amdgcn_wmma_f32_16x16x4_f32(
                false, afA[kk], false, bf1, (short)0, accA1, false, false);
            accB1 = __builtin_amdgcn_wmma_f32_16x16x4_f32(
                false, afB[kk], false, bf1, (short)0, accB1, false, false);
        }

        // distance proxy: ||w||^2 - 2 x.w  (||x||^2 constant per row);
        // n0 checked before n1 with strict '<' keeps earliest-index ties.
        const float wn0 = wn_lds[j0 + lm];
        const float wn1 = wn_lds[j0 + 16 + lm];
        const int n0 = j0 + lm, n1 = j0 + 16 + lm;
        #pragma unroll
        for (int r = 0; r < 8; ++r) {
            float dA0 = wn0 - 2.0f * accA0[r];
            if (dA0 < bestA[r]) { bestA[r] = dA0; bidxA[r] = n0; }
            float dA1 = wn1 - 2.0f * accA1[r];
            if (dA1 < bestA[r]) { bestA[r] = dA1; bidxA[r] = n1; }
            float dB0 = wn0 - 2.0f * accB0[r];
            if (dB0 < bestB[r]) { bestB[r] = dB0; bidxB[r] = n0; }
            float dB1 = wn1 - 2.0f * accB1[r];
            if (dB1 < bestB[r]) { bestB[r] = dB1; bidxB[r] = n1; }
        }
    }

    // ---- butterfly reduce over N within each 16-lane half (wave32) ----
    #pragma unroll
    for (int m = 1; m <= 8; m <<= 1) {
        #pragma unroll
        for (int r = 0; r < 8; ++r) {
            float ov = __shfl_xor(bestA[r], m, 32);
            int   oi = __shfl_xor(bidxA[r], m, 32);
            if (ov < bestA[r] || (ov == bestA[r] && oi < bidxA[r])) {
                bestA[r] = ov; bidxA[r] = oi;
            }
            float pv = __shfl_xor(bestB[r], m, 32);
            int   pi = __shfl_xor(bidxB[r], m, 32);
            if (pv < bestB[r] || (pv == bestB[r] && pi < bidxB[r])) {
                bestB[r] = pv; bidxB[r] = pi;
            }
        }
    }
    // C/D layout: VGPR r -> row r (lanes 0-15) / row r+8 (lanes 16-31)
    if (lm == 0) {
        #pragma unroll
        for (int r = 0; r < 8; ++r) {
            int rowA = tblA + 8 * lh + r;
            idx_lds[rowA] = bidxA[r];
            out[p0 + rowA] = (float)bidxA[r];   // discrete_latent
            int rowB = tblB + 8 * lh + r;
            idx_lds[rowB] = bidxB[r];
            out[p0 + rowB] = (float)bidxB[r];
        }
    }
    __syncthreads();

    // ---- gather quantized, write NCHW (coalesced per channel plane), loss ----
    float* outq = out + Q_OFF;
    const int pl  = t;                      // one pixel per thread
    const int idp = idx_lds[pl];
    const float* wq = w_lds + idp * WROW;
    const float* xp = xT_lds + pl * WROW;
    float* oq = outq + (size_t)b * (DIM * 4096) + hw0 + pl;

    float lsum = 0.f;
    #pragma unroll 8
    for (int c = 0; c < DIM; ++c) {
        float q  = wq[c];
        float df = q - xp[c];
        lsum += df * df;
        oq[(size_t)c * 4096] = q;   // straight-through output == quantized
    }

    // ---- deterministic block reduction of loss partial ----
    #pragma unroll
    for (int m = 1; m < 32; m <<= 1) lsum += __shfl_xor(lsum, m, 32);
    if (lane == 0) red[wave] = lsum;
    __syncthreads();
    if (t == 0) {
        float s = 0.f;
        #pragma unroll
        for (int i = 0; i < 8; ++i) s += red[i];
        partial[blockIdx.x] = s;
    }
}

__global__ __launch_bounds__(256)
void vq_loss(const float* __restrict__ partial, float* __restrict__ out)
{
    __shared__ float sm[256];
    float s = 0.f;
    for (int i = threadIdx.x; i < N_BLOCKS; i += 256) s += partial[i];
    sm[threadIdx.x] = s;
    __syncthreads();
    for (int m = 128; m > 0; m >>= 1) {
        if (threadIdx.x < m) sm[threadIdx.x] += sm[threadIdx.x + m];
        __syncthreads();
    }
    if (threadIdx.x == 0)
        out[Q_OFF + Q_ELEMS] = 1.25f * sm[0] / (float)Q_ELEMS;  // (1+0.25)*mse
}

extern "C" void kernel_launch(void* const* d_in, const int* in_sizes, int n_in,
                              void* d_out, int out_size, void* d_ws, size_t ws_size,
                              hipStream_t stream)
{
    const float* x = (const float*)d_in[0];
    const float* w = (const float*)d_in[1];
    float* out = (float*)d_out;
    float* partial = (float*)d_ws;   // N_BLOCKS floats

    const size_t lds_bytes =
        (size_t)(K_CODES * WROW + PIX_PER_BLOCK * WROW + K_CODES) * sizeof(float)
        + PIX_PER_BLOCK * sizeof(int) + 8 * sizeof(float);

    vq_main<<<N_BLOCKS, THREADS, lds_bytes, stream>>>(x, w, out, partial);
    vq_loss<<<1, 256, 0, stream>>>(partial, out);
}